// NTupleApproximator_3667902071457
// MI455X (gfx1250) — compile-verified
//
#include <hip/hip_runtime.h>
#include <stdint.h>

#define BOARDS_PER_BLOCK 256
#define NTUP 32
#define CELLS 16

__global__ __launch_bounds__(BOARDS_PER_BLOCK)
void ntuple_eval_kernel(const int* __restrict__ boards,
                        const float* __restrict__ weights,
                        const int* __restrict__ sym_coords,
                        const int* __restrict__ table_ids,
                        float* __restrict__ out,
                        int B)
{
    // 256 boards * 16 cells * 4B = 16 KB staged per block
    __shared__ __align__(16) uint32_t lds_boards[BOARDS_PER_BLOCK * CELLS];

    const unsigned t = threadIdx.x;
    const unsigned blockBase = blockIdx.x * BOARDS_PER_BLOCK;

    uint64_t pack = 0;  // 16 cells, 4 bits each (cell values are 0..15)

    if (blockBase + BOARDS_PER_BLOCK <= (unsigned)B) {
        // ---- CDNA5 async tensor-path staging: fully coalesced 128-bit
        // global -> LDS DMA, tracked by ASYNCcnt (no VGPR round trip). ----
        const uint64_t gbase = (uint64_t)(uintptr_t)(boards + (size_t)blockBase * CELLS);
        const uint32_t lbase = (uint32_t)(uintptr_t)(void*)lds_boards; // low 32 bits = LDS offset
        #pragma unroll
        for (int k = 0; k < 4; ++k) {
            unsigned elem = t + k * BOARDS_PER_BLOCK;          // uint4 element index
            uint64_t gaddr = gbase + (uint64_t)elem * 16u;
            uint32_t laddr = lbase + elem * 16u;
            asm volatile("global_load_async_to_lds_b128 %0, %1, off"
                         :: "v"(laddr), "v"(gaddr) : "memory");
        }
        asm volatile("s_wait_asynccnt 0" ::: "memory");
        __syncthreads();

        const uint32_t* my = &lds_boards[t * CELLS];
        #pragma unroll
        for (int k = 0; k < 4; ++k) {
            uint4 q = *(const uint4*)(my + 4 * k);
            pack |= (uint64_t)(q.x & 0xFu) << (4 * (4 * k + 0));
            pack |= (uint64_t)(q.y & 0xFu) << (4 * (4 * k + 1));
            pack |= (uint64_t)(q.z & 0xFu) << (4 * (4 * k + 2));
            pack |= (uint64_t)(q.w & 0xFu) << (4 * (4 * k + 3));
        }
    } else {
        // Tail block fallback: direct guarded loads
        unsigned b = blockBase + t;
        if (b < (unsigned)B) {
            const int* cells = boards + (size_t)b * CELLS;
            #pragma unroll
            for (int i = 0; i < CELLS; ++i)
                pack |= (uint64_t)((unsigned)cells[i] & 0xFu) << (4 * i);
        }
    }

    const unsigned b = blockBase + t;
    if (b >= (unsigned)B) return;

    // Phase 1: all 32 feature indices (sym_coords reads are wave-uniform
    // -> scalar loads; shifts use scalar shift amounts).
    unsigned feats[NTUP];
    #pragma unroll
    for (int tu = 0; tu < NTUP; ++tu) {
        unsigned f = 0;
        #pragma unroll
        for (int j = 0; j < 6; ++j) {
            int c = sym_coords[tu * 6 + j];                 // uniform
            f = (f << 4) | (unsigned)((pack >> (4 * c)) & 0xFull);
        }
        feats[tu] = f;   // 24-bit index, matches powers 16^5..16^0
    }

    // Phase 2: 32 independent gathers (L2-resident tables), 4 accumulator
    // chains so the scheduler can keep many loads in flight.
    float a0 = 0.f, a1 = 0.f, a2 = 0.f, a3 = 0.f;
    #pragma unroll
    for (int tu = 0; tu < NTUP; tu += 4) {
        const float* t0 = weights + ((size_t)table_ids[tu + 0] << 24); // uniform base
        const float* t1 = weights + ((size_t)table_ids[tu + 1] << 24);
        const float* t2 = weights + ((size_t)table_ids[tu + 2] << 24);
        const float* t3 = weights + ((size_t)table_ids[tu + 3] << 24);
        a0 += t0[feats[tu + 0]];
        a1 += t1[feats[tu + 1]];
        a2 += t2[feats[tu + 2]];
        a3 += t3[feats[tu + 3]];
    }
    out[b] = (a0 + a1) + (a2 + a3);
}

extern "C" void kernel_launch(void* const* d_in, const int* in_sizes, int n_in,
                              void* d_out, int out_size, void* d_ws, size_t ws_size,
                              hipStream_t stream) {
    const int*   boards     = (const int*)d_in[0];
    const float* weights    = (const float*)d_in[1];
    const int*   sym_coords = (const int*)d_in[2];
    const int*   table_ids  = (const int*)d_in[3];
    float*       out        = (float*)d_out;

    const int B = in_sizes[0] / CELLS;                 // 65536
    const int grid = (B + BOARDS_PER_BLOCK - 1) / BOARDS_PER_BLOCK;

    ntuple_eval_kernel<<<grid, BOARDS_PER_BLOCK, 0, stream>>>(
        boards, weights, sym_coords, table_ids, out, B);
}